// TACTiCSNet_81518479278510
// MI455X (gfx1250) — compile-verified
//
#include <hip/hip_runtime.h>
#include <hip/hip_bf16.h>
#include <math.h>

// ---------------- problem constants (match reference) ----------------
#define N_ROWS   4096          // N_A == N_B
#define G_DIM    4000          // GA == GB
#define EMB      32            // LEN_EMB
#define FIN      16            // LEN_FIN
#define NCA      50
#define NCB      60
#define KNN      10
#define N_ALL    (2 * N_ROWS)  // 8192

typedef __attribute__((ext_vector_type(2))) float f32x2;
typedef __attribute__((ext_vector_type(8))) float f32x8;

// =====================================================================
// WMMA GEMM #1: Ws = (GM @ Wk + Wadd) / std      (G x 32)
//   GM:   G x K   (row major, K = 4000)
//   Wk:   K x 32  (row major)
//   Wadd: G x 32, std: G
// One wave per 16x32 output tile; V_WMMA_F32_16X16X4_F32, K step 4.
// A layout: lane l -> M = l&15, K = 2*(l>>4) + v     (v = component)
// B layout: lane l -> N = l&15 (+16 for 2nd C tile), K = 2*(l>>4) + v
// C layout: vgpr r, lane l -> M = r + 8*(l>>4), N = l&15
// =====================================================================
__global__ void weff_kernel(const float* __restrict__ GM,
                            const float* __restrict__ Wk,
                            const float* __restrict__ Wadd,
                            const float* __restrict__ stdv,
                            float* __restrict__ Ws,
                            int G, int K)
{
    const int lane = threadIdx.x & 31;
    const int wave = threadIdx.x >> 5;
    const int tile = blockIdx.x * (blockDim.x >> 5) + wave;
    const int m0   = tile * 16;
    if (m0 >= G) return;                       // wave-uniform: EXEC stays all-ones

    const int half = lane >> 4;                // 0 or 1
    const int mr   = lane & 15;                // M (for A) / N (for B,C)

    f32x8 c0 = {}; f32x8 c1 = {};
    const float* arow = GM + (size_t)(m0 + mr) * K + 2 * half;

    for (int k0 = 0; k0 < K; k0 += 4) {
        f32x2 a = *(const f32x2*)(arow + k0);                 // A[m][k0+2h .. +1], 8B aligned
        const float* bp = Wk + (size_t)(k0 + 2 * half) * EMB + mr;
        f32x2 b0; b0.x = bp[0];  b0.y = bp[EMB];              // N = mr
        f32x2 b1; b1.x = bp[16]; b1.y = bp[EMB + 16];         // N = mr + 16
        c0 = __builtin_amdgcn_wmma_f32_16x16x4_f32(false, a, false, b0, (short)0, c0, false, false);
        c1 = __builtin_amdgcn_wmma_f32_16x16x4_f32(false, a, false, b1, (short)0, c1, false, false);
    }

#pragma unroll
    for (int r = 0; r < 8; ++r) {
        const int m = m0 + r + 8 * half;
        const float inv = 1.0f / stdv[m];
        Ws[(size_t)m * EMB + mr]      = (c0[r] + Wadd[(size_t)m * EMB + mr])      * inv;
        Ws[(size_t)m * EMB + mr + 16] = (c1[r] + Wadd[(size_t)m * EMB + mr + 16]) * inv;
    }
}

// b_eff[e] = embed_b[e] - sum_g mean[g] * Ws[g][e]     (32 threads)
__global__ void beff_kernel(const float* __restrict__ embed_b,
                            const float* __restrict__ mean,
                            const float* __restrict__ Ws,
                            float* __restrict__ beff, int G)
{
    const int e = threadIdx.x;
    if (e >= EMB) return;
    float s = embed_b[e];
    for (int g = 0; g < G; ++g) s -= mean[g] * Ws[(size_t)g * EMB + e];
    beff[e] = s;
}

// =====================================================================
// WMMA GEMM #2: Xout = relu(X @ Ws + beff)   (M x 32), M = 4096
// =====================================================================
__global__ void embed_kernel(const float* __restrict__ X,
                             const float* __restrict__ Ws,
                             const float* __restrict__ beff,
                             float* __restrict__ Xout,
                             int M, int K)
{
    const int lane = threadIdx.x & 31;
    const int wave = threadIdx.x >> 5;
    const int tile = blockIdx.x * (blockDim.x >> 5) + wave;
    const int m0   = tile * 16;
    if (m0 >= M) return;

    const int half = lane >> 4;
    const int mr   = lane & 15;

    f32x8 c0 = {}; f32x8 c1 = {};
    const float* arow = X + (size_t)(m0 + mr) * K + 2 * half;

    for (int k0 = 0; k0 < K; k0 += 4) {
        f32x2 a = *(const f32x2*)(arow + k0);
        const float* bp = Ws + (size_t)(k0 + 2 * half) * EMB + mr;
        f32x2 b0; b0.x = bp[0];  b0.y = bp[EMB];
        f32x2 b1; b1.x = bp[16]; b1.y = bp[EMB + 16];
        c0 = __builtin_amdgcn_wmma_f32_16x16x4_f32(false, a, false, b0, (short)0, c0, false, false);
        c1 = __builtin_amdgcn_wmma_f32_16x16x4_f32(false, a, false, b1, (short)0, c1, false, false);
    }

    const float bb0 = beff[mr];
    const float bb1 = beff[mr + 16];
#pragma unroll
    for (int r = 0; r < 8; ++r) {
        const int m = m0 + r + 8 * half;
        Xout[(size_t)m * EMB + mr]      = fmaxf(c0[r] + bb0, 0.0f);
        Xout[(size_t)m * EMB + mr + 16] = fmaxf(c1[r] + bb1, 0.0f);
    }
}

// =====================================================================
// Classifier head: preds = softmax(relu(x @ W1 + b1) @ W2 + b2)
// One thread per row; all loops fully unrolled (NC compile-time).
// =====================================================================
template <int NC>
__global__ void head_kernel(const float* __restrict__ X,
                            const float* __restrict__ W1, const float* __restrict__ b1,
                            const float* __restrict__ W2, const float* __restrict__ b2,
                            float* __restrict__ out, int nrows)
{
    const int i = blockIdx.x * blockDim.x + threadIdx.x;
    if (i >= nrows) return;

    float x[EMB];
#pragma unroll
    for (int e = 0; e < EMB; ++e) x[e] = X[(size_t)i * EMB + e];

    float h[FIN];
#pragma unroll
    for (int f = 0; f < FIN; ++f) {
        float s = b1[f];
#pragma unroll
        for (int e = 0; e < EMB; ++e) s += x[e] * W1[e * FIN + f];
        h[f] = fmaxf(s, 0.0f);
    }

    float lg[NC];
    float mx = -3.4e38f;
#pragma unroll
    for (int c = 0; c < NC; ++c) {
        float s = b2[c];
#pragma unroll
        for (int f = 0; f < FIN; ++f) s += h[f] * W2[f * NC + c];
        lg[c] = s;
        mx = fmaxf(mx, s);
    }
    float sum = 0.0f;
#pragma unroll
    for (int c = 0; c < NC; ++c) { float e = expf(lg[c] - mx); lg[c] = e; sum += e; }
    const float inv = 1.0f / sum;
#pragma unroll
    for (int c = 0; c < NC; ++c) out[(size_t)i * NC + c] = lg[c] * inv;
}

// ---------------- squared norms of all 8192 embedding rows ----------------
__global__ void sqn_kernel(const float* __restrict__ emb, float* __restrict__ sqn)
{
    const int i = blockIdx.x * blockDim.x + threadIdx.x;
    if (i >= N_ALL) return;
    const float* p = emb + (size_t)i * EMB;
    float s = 0.0f;
#pragma unroll
    for (int e = 0; e < EMB; ++e) s += p[e] * p[e];
    sqn[i] = s;
}

// =====================================================================
// kNN-mean over one 4096-column block. One wave per row.
// Each lane keeps an 11-entry sorted min-list (static indexing only),
// then the wave extracts the global 10/11 smallest via shfl_xor min +
// ballot-leader pop (leader shifts its list so head stays at best[0]).
// skip_first reproduces top_k(k+1)[:,1:] (drop single smallest = self).
// =====================================================================
__device__ __forceinline__ float knn_block(const float* __restrict__ emb,
                                           const float* __restrict__ sqn,
                                           const float4* __restrict__ ri,
                                           float sqi, int cbase, int lane,
                                           bool skip_first)
{
    const float BIG = 3.4e38f;
    float best[11];
#pragma unroll
    for (int t = 0; t < 11; ++t) best[t] = BIG;

    for (int j = cbase + lane; j < cbase + N_ROWS; j += 32) {
        const float4* rj = (const float4*)(emb + (size_t)j * EMB);
        float dot = 0.0f;
#pragma unroll
        for (int q = 0; q < 8; ++q) {
            float4 b = rj[q];
            dot += ri[q].x * b.x; dot += ri[q].y * b.y;
            dot += ri[q].z * b.z; dot += ri[q].w * b.w;
        }
        float sq = sqi + sqn[j] - 2.0f * dot;
        sq = fmaxf(sq, 0.0f);
        float d = (sq > 0.0f) ? sqrtf(sq) : 0.0f;
        if (d < best[10]) {            // sorted insert via compare-exchange sweep
            float v = d;
#pragma unroll
            for (int t = 0; t < 11; ++t) {
                float lo = fminf(best[t], v);
                float hi = fmaxf(best[t], v);
                best[t] = lo; v = hi;
            }
        }
    }

    const int cnt = skip_first ? (KNN + 1) : KNN;
    float sum = 0.0f;
    for (int t = 0; t < cnt; ++t) {
        float c = best[0];
        float m = c;
#pragma unroll
        for (int off = 16; off > 0; off >>= 1) m = fminf(m, __shfl_xor(m, off, 32));
        unsigned long long mask = __ballot(c == m);
        int leader = __ffsll(mask) - 1;
        if (lane == leader) {          // pop head: shift (static indices only)
#pragma unroll
            for (int t2 = 0; t2 < 10; ++t2) best[t2] = best[t2 + 1];
            best[10] = BIG;
        }
        if (!(skip_first && t == 0)) sum += m;
    }
    return sum * (1.0f / (float)KNN);
}

__global__ void __launch_bounds__(32)
knn_kernel(const float* __restrict__ emb, const float* __restrict__ sqn,
           float* __restrict__ rvals)
{
    const int i    = blockIdx.x;      // row 0..8191
    const int lane = threadIdx.x;
    const float* rip = emb + (size_t)i * EMB;
    float4 ri[8];
#pragma unroll
    for (int q = 0; q < 8; ++q) ri[q] = ((const float4*)rip)[q];
    const float sqi = sqn[i];

    const int selfBase  = (i < N_ROWS) ? 0 : N_ROWS;
    const int crossBase = (i < N_ROWS) ? N_ROWS : 0;

    float selfMean  = knn_block(emb, sqn, ri, sqi, selfBase,  lane, true);
    float crossMean = knn_block(emb, sqn, ri, sqi, crossBase, lane, false);

    if (lane == 0) {
        selfMean  = (selfMean  == 0.0f) ? 1.0f : selfMean;
        crossMean = (crossMean == 0.0f) ? 1.0f : crossMean;
        rvals[i] = crossMean / selfMean;
    }
}

// deterministic tree reduction -> mean -> relu -> scalar loss
__global__ void reduce_kernel(const float* __restrict__ rvals, float* __restrict__ out)
{
    __shared__ float sh[256];
    const int t = threadIdx.x;
    float s = 0.0f;
    for (int i = t; i < N_ALL; i += 256) s += rvals[i];
    sh[t] = s;
    __syncthreads();
    for (int st = 128; st > 0; st >>= 1) {
        if (t < st) sh[t] += sh[t + st];
        __syncthreads();
    }
    if (t == 0) out[0] = fmaxf(sh[0] / (float)N_ALL, 0.0f);
}

// =====================================================================
// launch
// =====================================================================
extern "C" void kernel_launch(void* const* d_in, const int* in_sizes, int n_in,
                              void* d_out, int out_size, void* d_ws, size_t ws_size,
                              hipStream_t stream)
{
    (void)in_sizes; (void)n_in; (void)out_size; (void)ws_size;

    const float* x_init_A = (const float*)d_in[0];
    const float* x_init_B = (const float*)d_in[1];
    const float* mean_A   = (const float*)d_in[2];
    const float* mean_B   = (const float*)d_in[3];
    const float* std_A    = (const float*)d_in[4];
    const float* std_B    = (const float*)d_in[5];
    const float* gmA      = (const float*)d_in[6];
    const float* gmB      = (const float*)d_in[7];
    const float* embed_W  = (const float*)d_in[8];   // (GA+GB) x 32
    const float* embed_b  = (const float*)d_in[9];
    const float* clsA_W1  = (const float*)d_in[10];
    const float* clsA_b1  = (const float*)d_in[11];
    const float* clsA_W2  = (const float*)d_in[12];
    const float* clsA_b2  = (const float*)d_in[13];
    const float* clsB_W1  = (const float*)d_in[14];
    const float* clsB_b1  = (const float*)d_in[15];
    const float* clsB_W2  = (const float*)d_in[16];
    const float* clsB_b2  = (const float*)d_in[17];

    float* out = (float*)d_out;
    // d_out layout: preds_A | preds_B | x_A | x_B | loss
    const size_t oPA   = 0;
    const size_t oPB   = oPA + (size_t)N_ROWS * NCA;
    const size_t oXA   = oPB + (size_t)N_ROWS * NCB;
    const size_t oXB   = oXA + (size_t)N_ROWS * EMB;
    const size_t oLoss = oXB + (size_t)N_ROWS * EMB;
    float* predsA = out + oPA;
    float* predsB = out + oPB;
    float* xA     = out + oXA;
    float* xB     = out + oXB;        // contiguous after xA -> 8192x32 table
    float* lossp  = out + oLoss;

    // workspace layout (floats)
    float* ws    = (float*)d_ws;
    float* WsA   = ws;                          // 4000*32
    float* WsB   = WsA  + (size_t)G_DIM * EMB;  // 4000*32
    float* beffA = WsB  + (size_t)G_DIM * EMB;  // 32
    float* beffB = beffA + EMB;                 // 32
    float* sqn   = beffB + EMB;                 // 8192
    float* rvals = sqn + N_ALL;                 // 8192

    const float* embW_A = embed_W;                       // rows [0, GA)
    const float* embW_B = embed_W + (size_t)G_DIM * EMB; // rows [GA, GA+GB)

    // ---- effective weights (WMMA f32 GEMM, 4000x4000x32) ----
    {
        const int tiles  = G_DIM / 16;                   // 250
        const int wavesB = 8;                            // 256-thread blocks
        const int blocks = (tiles + wavesB - 1) / wavesB;
        weff_kernel<<<blocks, 256, 0, stream>>>(gmA, embW_B, embW_A, std_A, WsA, G_DIM, G_DIM);
        weff_kernel<<<blocks, 256, 0, stream>>>(gmB, embW_A, embW_B, std_B, WsB, G_DIM, G_DIM);
    }
    beff_kernel<<<1, EMB, 0, stream>>>(embed_b, mean_A, WsA, beffA, G_DIM);
    beff_kernel<<<1, EMB, 0, stream>>>(embed_b, mean_B, WsB, beffB, G_DIM);

    // ---- embeddings (WMMA f32 GEMM, 4096x4000x32 + bias + relu) ----
    {
        const int tiles  = N_ROWS / 16;                  // 256
        const int blocks = (tiles + 7) / 8;
        embed_kernel<<<blocks, 256, 0, stream>>>(x_init_A, WsA, beffA, xA, N_ROWS, G_DIM);
        embed_kernel<<<blocks, 256, 0, stream>>>(x_init_B, WsB, beffB, xB, N_ROWS, G_DIM);
    }

    // ---- classifier heads + softmax ----
    head_kernel<NCA><<<(N_ROWS + 255) / 256, 256, 0, stream>>>(xA, clsA_W1, clsA_b1, clsA_W2, clsA_b2, predsA, N_ROWS);
    head_kernel<NCB><<<(N_ROWS + 255) / 256, 256, 0, stream>>>(xB, clsB_W1, clsB_b1, clsB_W2, clsB_b2, predsB, N_ROWS);

    // ---- embedding loss ----
    sqn_kernel<<<(N_ALL + 255) / 256, 256, 0, stream>>>(xA /* 8192x32 table */, sqn);
    knn_kernel<<<N_ALL, 32, 0, stream>>>(xA, sqn, rvals);
    reduce_kernel<<<1, 256, 0, stream>>>(rvals, lossp);
}